// FGCN_48687749268219
// MI455X (gfx1250) — compile-verified
//
#include <hip/hip_runtime.h>
#include <hip/hip_bf16.h>

typedef __attribute__((ext_vector_type(2))) float v2f;
typedef __attribute__((ext_vector_type(8))) float v8f;

// ---------------------------------------------------------------------------
// Pre-pack W[K x 64] into WMMA B-fragment order:
//   frag (kk = k/4, n) for lane L is the float2
//     { W[k + (L>=16?2:0)    , n*16 + (L&15)],
//       W[k + (L>=16?2:0) + 1, n*16 + (L&15)] }
// stored contiguously at Wp + ((kk*4 + n)*32 + L)*2.  One thread per float2.
// ---------------------------------------------------------------------------
__global__ __launch_bounds__(256) void fgcn_pack_w(const float* __restrict__ W,
                                                   float* __restrict__ Wp,
                                                   int total /* = (K/4)*4*32 */) {
    int tid = blockIdx.x * blockDim.x + threadIdx.x;
    if (tid >= total) return;
    const int lane = tid & 31;
    const int n    = (tid >> 5) & 3;
    const int kk   = tid >> 7;
    const int kb   = kk * 4 + ((lane >= 16) ? 2 : 0);
    const int col  = n * 16 + (lane & 15);
    float2 v;
    v.x = W[(size_t)kb * 64 + col];
    v.y = W[(size_t)(kb + 1) * 64 + col];
    *(float2*)(Wp + (size_t)tid * 2) = v;
}

// ---------------------------------------------------------------------------
// Dense GEMM  H[M x 64] = X[M x K] @ W[K x 64]  using V_WMMA_F32_16X16X4_F32.
// One wave per 16-row tile; 4 accumulators cover all 64 output columns.
// Per k-step: 1 global b64 (A frag) + 4 global b64 (pre-packed B frags) + 4 WMMA.
// M multiple of 16, K multiple of 4 (true here).
// ---------------------------------------------------------------------------
__global__ __launch_bounds__(256) void fgcn_gemm_wmma(const float* __restrict__ X,
                                                      const float* __restrict__ Wp,
                                                      float* __restrict__ H,
                                                      int mtiles, int K) {
    const int wave = (blockIdx.x * blockDim.x + threadIdx.x) >> 5;  // wave-uniform
    const int lane = threadIdx.x & 31;
    if (wave >= mtiles) return;  // uniform branch: EXEC stays all-ones for WMMA

    const int m0     = wave * 16;
    const int lane16 = lane & 15;
    const bool hi    = lane >= 16;

    v8f acc0 = {}, acc1 = {}, acc2 = {}, acc3 = {};

    // A fragment source: lane-dependent K offset (+2 for upper half-wave).
    const float* xa = X + (size_t)(m0 + lane16) * K + (hi ? 2 : 0);
    // B fragment source: packed stream, 4 frags (n=0..3) of 32 lanes per k-step.
    const float* wb = Wp + (size_t)lane * 2;

    for (int k = 0; k < K; k += 4) {
        float2 af = *(const float2*)(xa + k);
        v2f a; a.x = af.x; a.y = af.y;

        const float* wk = wb + (size_t)(k >> 2) * 256;  // 4 frags * 32 lanes * 2
        v2f b0 = *(const v2f*)(wk);
        v2f b1 = *(const v2f*)(wk + 64);
        v2f b2 = *(const v2f*)(wk + 128);
        v2f b3 = *(const v2f*)(wk + 192);

        acc0 = __builtin_amdgcn_wmma_f32_16x16x4_f32(false, a, false, b0, (short)0, acc0, false, false);
        acc1 = __builtin_amdgcn_wmma_f32_16x16x4_f32(false, a, false, b1, (short)0, acc1, false, false);
        acc2 = __builtin_amdgcn_wmma_f32_16x16x4_f32(false, a, false, b2, (short)0, acc2, false, false);
        acc3 = __builtin_amdgcn_wmma_f32_16x16x4_f32(false, a, false, b3, (short)0, acc3, false, false);
    }

    // C/D layout: VGPR r -> row m0+r (lanes 0-15) / m0+8+r (lanes 16-31), col = lane&15.
    const int rbase = m0 + (hi ? 8 : 0);
    #pragma unroll
    for (int r = 0; r < 8; ++r) {
        float* o = H + (size_t)(rbase + r) * 64 + lane16;
        o[0]  = acc0[r];
        o[16] = acc1[r];
        o[32] = acc2[r];
        o[48] = acc3[r];
    }
}

// ---------------------------------------------------------------------------
// out[i, f] = bias[f]  (initialize accumulation target)
// ---------------------------------------------------------------------------
__global__ __launch_bounds__(256) void fgcn_bias_init(float* __restrict__ out,
                                                      const float* __restrict__ bias,
                                                      int total) {
    int idx = blockIdx.x * blockDim.x + threadIdx.x;
    if (idx < total) out[idx] = bias[idx & 63];
}

// ---------------------------------------------------------------------------
// Edge scatter: out[dst[e], :] += h[src[e], :]; 16 threads per edge, float4 each.
// ---------------------------------------------------------------------------
__global__ __launch_bounds__(256) void fgcn_edge_scatter(const float* __restrict__ h,
                                                         const int* __restrict__ src,
                                                         const int* __restrict__ dst,
                                                         float* __restrict__ out,
                                                         int E) {
    int idx = blockIdx.x * blockDim.x + threadIdx.x;
    if (idx >= E * 16) return;
    const int e = idx >> 4;
    const int q = (idx & 15) << 2;  // feature offset 0..60 step 4
    const int s = src[e];
    const int d = dst[e];
    float4 v = *(const float4*)(h + (size_t)s * 64 + q);
    float* o = out + (size_t)d * 64 + q;
    atomicAdd(o + 0, v.x);
    atomicAdd(o + 1, v.y);
    atomicAdd(o + 2, v.z);
    atomicAdd(o + 3, v.w);
}

// ---------------------------------------------------------------------------
// Launch
// ---------------------------------------------------------------------------
static constexpr int N_DRUG = 50000, E_DRUG = 1000000, F_DRUG = 256;
static constexpr int N_DIS  = 20000, E_DIS  = 640000,  F_DIS  = 128;
static constexpr int NHID2  = 64;

extern "C" void kernel_launch(void* const* d_in, const int* in_sizes, int n_in,
                              void* d_out, int out_size, void* d_ws, size_t ws_size,
                              hipStream_t stream) {
    const float* drug_x  = (const float*)d_in[0];
    const float* dis_x   = (const float*)d_in[1];
    const float* W1      = (const float*)d_in[2];
    const float* b1      = (const float*)d_in[3];
    const float* W2      = (const float*)d_in[4];
    const float* b2      = (const float*)d_in[5];
    const int*   drug_ei = (const int*)d_in[6];  // [2, E_DRUG] flat: src then dst
    const int*   dis_ei  = (const int*)d_in[7];  // [2, E_DIS]

    float* out1 = (float*)d_out;                       // emb1: [N_DRUG, 64]
    float* out2 = out1 + (size_t)N_DRUG * NHID2;       // emb2: [N_DIS, 64]

    float* h1  = (float*)d_ws;                         // [N_DRUG, 64] = 12.8 MB
    float* h2  = h1 + (size_t)N_DRUG * NHID2;          // [N_DIS, 64]  =  5.1 MB
    float* Wp1 = h2 + (size_t)N_DIS * NHID2;           // 256*64 floats = 64 KB
    float* Wp2 = Wp1 + (size_t)F_DRUG * NHID2;         // 128*64 floats = 32 KB

    // 0) Pre-pack weights into WMMA B-fragment order (tiny).
    {
        int total = F_DRUG * NHID2 / 2;                // (K/4)*4*32 float2's
        fgcn_pack_w<<<(total + 255) / 256, 256, 0, stream>>>(W1, Wp1, total);
    }
    {
        int total = F_DIS * NHID2 / 2;
        fgcn_pack_w<<<(total + 255) / 256, 256, 0, stream>>>(W2, Wp2, total);
    }

    // 1) Dense GEMMs into workspace (WMMA f32 16x16x4).
    {
        int mtiles = N_DRUG / 16;                      // 3125
        int grid = (mtiles + 7) / 8;                   // 8 waves/block
        fgcn_gemm_wmma<<<grid, 256, 0, stream>>>(drug_x, Wp1, h1, mtiles, F_DRUG);
    }
    {
        int mtiles = N_DIS / 16;                       // 1250
        int grid = (mtiles + 7) / 8;
        fgcn_gemm_wmma<<<grid, 256, 0, stream>>>(dis_x, Wp2, h2, mtiles, F_DIS);
    }

    // 2) Initialize outputs with bias (accumulation base).
    {
        int total = N_DRUG * NHID2;
        fgcn_bias_init<<<(total + 255) / 256, 256, 0, stream>>>(out1, b1, total);
    }
    {
        int total = N_DIS * NHID2;
        fgcn_bias_init<<<(total + 255) / 256, 256, 0, stream>>>(out2, b2, total);
    }

    // 3) Edge gather + scatter-add (atomics land in L2; outputs fit easily).
    {
        int threads = E_DRUG * 16;
        fgcn_edge_scatter<<<(threads + 255) / 256, 256, 0, stream>>>(
            h1, drug_ei, drug_ei + E_DRUG, out1, E_DRUG);
    }
    {
        int threads = E_DIS * 16;
        fgcn_edge_scatter<<<(threads + 255) / 256, 256, 0, stream>>>(
            h2, dis_ei, dis_ei + E_DIS, out2, E_DIS);
    }
}